// TokenMoERouter_9448928051671
// MI455X (gfx1250) — compile-verified
//
#include <hip/hip_runtime.h>

// MoE token router for MI455X (gfx1250, wave32).
// logits = x @ W^T via V_WMMA_F32_16X16X4_F32 (full fp32 matrix pipe).
// W k-chunks staged with GLOBAL_LOAD_ASYNC_TO_LDS_B128 (ASYNCcnt-tracked,
// no VGPR bounce) and double-buffered in LDS so staging overlaps WMMA.
// Fused softmax + top-8 + renormalize epilogue (logits never touch HBM).

typedef __attribute__((ext_vector_type(2))) float v2f;
typedef __attribute__((ext_vector_type(8))) float v8f;

#define D_MODEL 2048
#define N_GROUPS 64
#define KCHUNK 128
#define TOP_K 8
#define CHUNK_FLOATS (N_GROUPS * KCHUNK)       // 8192 floats = 32 KB

// Async-stage W[:, kc:kc+KCHUNK] into LDS at byte offset lds_dst.
// 2048 float4 lines, 256 threads -> 8 GLOBAL_LOAD_ASYNC_TO_LDS_B128 each.
__device__ __forceinline__ void stage_W_async(unsigned long long wbase, int kc,
                                              unsigned lds_dst, int tid) {
    #pragma unroll
    for (int i = 0; i < 8; ++i) {
        int idx4 = tid + i * 256;                        // float4 index 0..2047
        int row  = idx4 >> 5;                            // group 0..63
        int c4   = (idx4 & 31) << 2;                     // col 0..124 step 4
        unsigned voff = (unsigned)(((row * D_MODEL) + kc + c4) * 4); // bytes
        unsigned ldst = lds_dst + (unsigned)((row * KCHUNK + c4) * 4);
        asm volatile("global_load_async_to_lds_b128 %0, %1, %2"
                     :: "v"(ldst), "v"(voff), "s"(wbase)
                     : "memory");
    }
}

__device__ __forceinline__ void wait_async0() {
    asm volatile("s_wait_asynccnt 0x0" ::: "memory");
}

__global__ __launch_bounds__(256)
void moe_router_wmma(const float* __restrict__ x,
                     const float* __restrict__ W,
                     float* __restrict__ out,
                     int T) {
    // 64 KB LDS: two ping-pong W-chunk buffers [64 groups][KCHUNK];
    // epilogue reuses the first 32 KB for logits [128 tok][64 g].
    __shared__ float lds[2 * CHUNK_FLOATS];

    const int tid  = threadIdx.x;
    const int wave = tid >> 5;
    const int lane = tid & 31;
    const int lm   = lane & 15;   // M (A) / N (B) index within tile
    const int hi   = lane >> 4;   // half-wave selector (K pair)
    const int tok0 = blockIdx.x * 128 + wave * 16;

    const unsigned long long wbase = (unsigned long long)W;
    // Low 32 bits of a generic LDS pointer = allocation-relative LDS offset.
    const unsigned lds0 = (unsigned)(unsigned long long)(&lds[0]);

    v8f acc0 = {}, acc1 = {}, acc2 = {}, acc3 = {};

    // A fragment base: row = tok0+lm, K offset = 2*hi (+j within the b64 load)
    const float* aptr = x + (size_t)(tok0 + lm) * D_MODEL + 2 * hi;

    // Prologue: stage chunk 0 into buffer 0.
    stage_W_async(wbase, 0, lds0, tid);
    wait_async0();
    __syncthreads();

    int cur = 0;
    for (int kc = 0; kc < D_MODEL; kc += KCHUNK) {
        // Kick off async staging of the next chunk into the idle buffer;
        // it lands while we run WMMAs on the current buffer.
        if (kc + KCHUNK < D_MODEL)
            stage_W_async(wbase, kc + KCHUNK,
                          lds0 + (unsigned)((cur ^ 1) * CHUNK_FLOATS * 4), tid);

        const float* lbuf = &lds[cur * CHUNK_FLOATS];
        #pragma unroll 4
        for (int kk = 0; kk < KCHUNK; kk += 4) {
            v2f a  = *(const v2f*)(aptr + kc + kk);
            // B fragment (K=kk+2*hi+j, N=lm) for each of the 4 group tiles
            v2f b0 = *(const v2f*)(&lbuf[( 0 + lm) * KCHUNK + kk + 2 * hi]);
            v2f b1 = *(const v2f*)(&lbuf[(16 + lm) * KCHUNK + kk + 2 * hi]);
            v2f b2 = *(const v2f*)(&lbuf[(32 + lm) * KCHUNK + kk + 2 * hi]);
            v2f b3 = *(const v2f*)(&lbuf[(48 + lm) * KCHUNK + kk + 2 * hi]);
            acc0 = __builtin_amdgcn_wmma_f32_16x16x4_f32(false, a, false, b0, (short)0, acc0, false, false);
            acc1 = __builtin_amdgcn_wmma_f32_16x16x4_f32(false, a, false, b1, (short)0, acc1, false, false);
            acc2 = __builtin_amdgcn_wmma_f32_16x16x4_f32(false, a, false, b2, (short)0, acc2, false, false);
            acc3 = __builtin_amdgcn_wmma_f32_16x16x4_f32(false, a, false, b3, (short)0, acc3, false, false);
        }

        // My async stores must have landed before anyone reads the new buffer;
        // the barrier also guarantees all waves finished reading `cur` before
        // the next iteration re-stages into it.
        wait_async0();
        __syncthreads();
        cur ^= 1;
    }

    // Scatter C tiles to LDS token-major: C layout is lane L, vgpr r ->
    // (M = r + 8*(L/16), N = L%16).  (Post-loop barrier already done.)
    #pragma unroll
    for (int r = 0; r < 8; ++r) {
        int m = r + 8 * hi;
        float* row = &lds[(wave * 16 + m) * N_GROUPS + lm];
        row[ 0] = acc0[r];
        row[16] = acc1[r];
        row[32] = acc2[r];
        row[48] = acc3[r];
    }
    __syncthreads();

    // Fused softmax + top-8 + renorm: 128 threads, one token each.
    if (tid < 128) {
        float* tl = &lds[tid * N_GROUPS];

        float mx = -3.402823466e38f;
        #pragma unroll 8
        for (int g = 0; g < N_GROUPS; ++g) mx = fmaxf(mx, tl[g]);

        float s = 0.f;
        #pragma unroll 8
        for (int g = 0; g < N_GROUPS; ++g) {
            float e = __expf(tl[g] - mx);
            tl[g] = e;            // keep exp in LDS (monotonic with score)
            s += e;
        }

        unsigned sel_lo = 0u, sel_hi = 0u;
        float topsum = 0.f;
        #pragma unroll
        for (int it = 0; it < TOP_K; ++it) {
            float best = -1.f; int bi = 0;
            for (int g = 0; g < N_GROUPS; ++g) {
                unsigned ch = (g < 32) ? ((sel_lo >> g) & 1u)
                                       : ((sel_hi >> (g - 32)) & 1u);
                float e = tl[g];
                if (!ch && e > best) { best = e; bi = g; }
            }
            topsum += best;
            if (bi < 32) sel_lo |= (1u << bi); else sel_hi |= (1u << (bi - 32));
        }

        float inv_s = 1.f / s;
        float inv_d = 1.f / (topsum * inv_s + 1e-8f);  // 1/(sum(top8 scores)+EPS)
        size_t TG   = (size_t)T * N_GROUPS;
        size_t base = (size_t)(blockIdx.x * 128 + tid) * N_GROUPS;

        #pragma unroll 4
        for (int g = 0; g < N_GROUPS; ++g) {
            float sc = tl[g] * inv_s;
            unsigned ch = (g < 32) ? ((sel_lo >> g) & 1u)
                                   : ((sel_hi >> (g - 32)) & 1u);
            out[base + g]          = ch ? sc * inv_d : 0.f;  // routing_weights
            out[TG + base + g]     = ch ? 1.f : 0.f;         // active_mask
            out[2 * TG + base + g] = sc;                     // scores
        }
    }
}

extern "C" void kernel_launch(void* const* d_in, const int* in_sizes, int n_in,
                              void* d_out, int out_size, void* d_ws, size_t ws_size,
                              hipStream_t stream) {
    const float* x = (const float*)d_in[0];
    const float* W = (const float*)d_in[1];
    float* out = (float*)d_out;
    int T = in_sizes[0] / D_MODEL;            // 16384 tokens
    int blocks = T / 128;                     // 128 tokens per block
    moe_router_wmma<<<dim3(blocks), dim3(256), 0, stream>>>(x, W, out, T);
}